// PedGraphConvolution_5446018531666
// MI455X (gfx1250) — compile-verified
//
#include <hip/hip_runtime.h>
#include <hip/hip_bf16.h>

#define NB  64
#define TT  512
#define VVD 17
#define OO  64
#define TV  (TT*VVD)        // 8704
#define NTV (NB*TV)         // 557056
#define NT  (NB*TT)         // 32768
#define EPSF 1e-5f

typedef __attribute__((ext_vector_type(2))) float v2f;
typedef __attribute__((ext_vector_type(8))) float v8f;

// workspace layout (float offsets)
#define WS_A1 0
#define WS_A2 (NTV)
#define WS_Y  (2*NTV)
#define WS_AM (5*NTV)            // 289 floats (A matrix), padded to 512
#define WS_ST (5*NTV + 512)     // 64 floats of stats/accumulators
#define WS_W  (WS_ST + 64)      // fused W: 8 x 64
#define WS_BI (WS_W + 512)      // fused bias: 64  (contiguous after W)

// ---- k1: a1/a2 = x . (w^T convT_w) + (b . convT_w) -------------------------
__global__ void k1_proj(const float* __restrict__ x,
                        const float* __restrict__ w1, const float* __restrict__ b1,
                        const float* __restrict__ w2, const float* __restrict__ b2,
                        const float* __restrict__ cw, const float* __restrict__ cbp,
                        float* __restrict__ a1, float* __restrict__ a2) {
  int idx = blockIdx.x * blockDim.x + threadIdx.x;
  if (idx >= NTV) return;
  float u1[3] = {0.f,0.f,0.f}, u2[3] = {0.f,0.f,0.f};
  float d1 = 0.f, d2 = 0.f;
  for (int c = 0; c < 3; ++c) {
    float wc = cw[c];
    d1 += wc * b1[c]; d2 += wc * b2[c];
    for (int k = 0; k < 3; ++k) { u1[k] += wc * w1[c*3+k]; u2[k] += wc * w2[c*3+k]; }
  }
  const float* xp = x + idx*3;
  float x0 = xp[0], x1 = xp[1], x2 = xp[2];
  a1[idx] = x0*u1[0] + x1*u1[1] + x2*u1[2] + d1;
  a2[idx] = x0*u2[0] + x1*u2[1] + x2*u2[2] + d2;
}

// ---- k2: separable mean/var of s over all N*T*V*V --------------------------
__global__ void k2_sstats(const float* __restrict__ a1, const float* __restrict__ a2,
                          const float* __restrict__ cbp, float* __restrict__ stats) {
  __shared__ float r1[256], r2[256];
  int nt = blockIdx.x * blockDim.x + threadIdx.x;
  float lsum = 0.f, lsq = 0.f;
  if (nt < NT) {
    float cb = cbp[0];
    const float* p1 = a1 + nt*VVD; const float* p2 = a2 + nt*VVD;
    float S1=0.f,S2=0.f,T1=0.f,T2=0.f;
    for (int v = 0; v < VVD; ++v) { float a=p1[v], b=p2[v]; S1+=a; S2+=a*a; T1+=b; T2+=b*b; }
    const float Vf = (float)VVD;
    lsum = Vf*(S1 - T1) + Vf*Vf*cb;
    lsq  = Vf*(S2 + T2) + Vf*Vf*cb*cb + 2.f*cb*Vf*(S1 - T1) - 2.f*S1*T1;
  }
  r1[threadIdx.x] = lsum; r2[threadIdx.x] = lsq; __syncthreads();
  for (int s = 128; s > 0; s >>= 1) {
    if (threadIdx.x < s) { r1[threadIdx.x] += r1[threadIdx.x+s]; r2[threadIdx.x] += r2[threadIdx.x+s]; }
    __syncthreads();
  }
  if (threadIdx.x == 0) { atomicAdd(&stats[0], r1[0]); atomicAdd(&stats[1], r2[0]); }
}

__global__ void k3_final(float* stats) {
  const float cnt = 9469952.f;      // N*T*V*V
  float mean = stats[0] / cnt;
  float var  = stats[1] / cnt - mean*mean;
  stats[2] = mean;
  stats[3] = rsqrtf(var + EPSF);
}

// ---- k4: a3 (temporal abs-diff of BN+ReLU'd s) folded into A ---------------
__global__ void k4_adj(const float* __restrict__ a1, const float* __restrict__ a2,
                       const float* __restrict__ cbp, const float* __restrict__ gp,
                       const float* __restrict__ btp, const float* __restrict__ stats,
                       const float* __restrict__ PA, const float* __restrict__ alpha,
                       float* __restrict__ AM) {
  __shared__ float red[64];
  int blk = blockIdx.x;              // i*V + j
  int i = blk / VVD, j = blk % VVD;
  int n = threadIdx.x;               // 64 threads = batch
  float cb = cbp[0], g = gp[0], bt = btp[0], m = stats[2], inv = stats[3];
  int base = n * TT * VVD;
  float s0 = a1[base + j] - a2[base + i] + cb;
  float prev = fmaxf(g*(s0 - m)*inv + bt, 0.f);
  float acc = 0.f;
  for (int t = 1; t < TT; ++t) {
    int o = base + t*VVD;
    float sc = a1[o + j] - a2[o + i] + cb;
    float cur = fmaxf(g*(sc - m)*inv + bt, 0.f);
    acc += fabsf(cur - prev);
    prev = cur;
  }
  red[n] = acc; __syncthreads();
  for (int st = 32; st > 0; st >>= 1) { if (n < st) red[n] += red[n+st]; __syncthreads(); }
  if (n == 0) AM[blk] = PA[blk] + alpha[blk] * (red[0] / (float)NB);
}

// ---- k5: y[n,c,t,:] = xr[n,c,t,:] @ A --------------------------------------
__global__ void k5_y(const float* __restrict__ xf, const float* __restrict__ AM,
                     float* __restrict__ y) {
  const int total = 3*NTV;
  for (int idx = blockIdx.x*blockDim.x + threadIdx.x; idx < total; idx += gridDim.x*blockDim.x) {
    int j = idx % VVD;
    int rb = idx - j;
    float s = 0.f;
    #pragma unroll
    for (int v = 0; v < VVD; ++v) s += xf[rb+v] * AM[v*VVD + j];
    y[idx] = s;
  }
}

// ---- k6: means + 3x3 second moments of y and xr (for analytic BN) ----------
__global__ void k6_moments(const float* __restrict__ y, const float* __restrict__ xf,
                           float* __restrict__ stats) {
  __shared__ float red[256];
  float acc[18];
  #pragma unroll
  for (int q = 0; q < 18; ++q) acc[q] = 0.f;
  for (int r = blockIdx.x*blockDim.x + threadIdx.x; r < NTV; r += gridDim.x*blockDim.x) {
    int n = r / TV, q = r % TV;
    int base = n*(3*TV) + q;
    float y0=y[base], y1=y[base+TV], y2=y[base+2*TV];
    float x0=xf[base], x1=xf[base+TV], x2=xf[base+2*TV];
    acc[0]+=y0; acc[1]+=y1; acc[2]+=y2;
    acc[3]+=y0*y0; acc[4]+=y0*y1; acc[5]+=y0*y2; acc[6]+=y1*y1; acc[7]+=y1*y2; acc[8]+=y2*y2;
    acc[9]+=x0; acc[10]+=x1; acc[11]+=x2;
    acc[12]+=x0*x0; acc[13]+=x0*x1; acc[14]+=x0*x2; acc[15]+=x1*x1; acc[16]+=x1*x2; acc[17]+=x2*x2;
  }
  for (int q = 0; q < 18; ++q) {
    red[threadIdx.x] = acc[q]; __syncthreads();
    for (int s = 128; s > 0; s >>= 1) { if (threadIdx.x < s) red[threadIdx.x] += red[threadIdx.x+s]; __syncthreads(); }
    if (threadIdx.x == 0) atomicAdd(&stats[4+q], red[0]);
    __syncthreads();
  }
}

// ---- k7: fold conv + BN (both branches) into W[8][64], bias[64] ------------
__global__ void k7_fuse(const float* __restrict__ stats,
                        const float* __restrict__ cw2, const float* __restrict__ cb2,
                        const float* __restrict__ bg, const float* __restrict__ bb,
                        const float* __restrict__ pw, const float* __restrict__ pb,
                        const float* __restrict__ pg, const float* __restrict__ pbeta,
                        float* __restrict__ Wf, float* __restrict__ bias) {
  (void)cb2; (void)pb;  // conv/pool biases cancel inside BatchNorm
  int o = threadIdx.x;
  if (o >= OO) return;
  const float invn = 1.f / (float)NTV;
  float my[3], mx[3], Ey[6], Ex[6];
  for (int c = 0; c < 3; ++c) { my[c] = stats[4+c]*invn; mx[c] = stats[13+c]*invn; }
  for (int q = 0; q < 6; ++q) { Ey[q] = stats[7+q]*invn; Ex[q] = stats[16+q]*invn; }
  float Cy[6] = { Ey[0]-my[0]*my[0], Ey[1]-my[0]*my[1], Ey[2]-my[0]*my[2],
                  Ey[3]-my[1]*my[1], Ey[4]-my[1]*my[2], Ey[5]-my[2]*my[2] };
  float Cx[6] = { Ex[0]-mx[0]*mx[0], Ex[1]-mx[0]*mx[1], Ex[2]-mx[0]*mx[2],
                  Ex[3]-mx[1]*mx[1], Ex[4]-mx[1]*mx[2], Ex[5]-mx[2]*mx[2] };
  float wz0=cw2[o*3+0], wz1=cw2[o*3+1], wz2=cw2[o*3+2];
  float varz = wz0*wz0*Cy[0] + wz1*wz1*Cy[3] + wz2*wz2*Cy[5]
             + 2.f*(wz0*wz1*Cy[1] + wz0*wz2*Cy[2] + wz1*wz2*Cy[4]);
  float sZ = bg[o] * rsqrtf(varz + EPSF);
  float wp0=pw[o*3+0], wp1=pw[o*3+1], wp2=pw[o*3+2];
  float varp = wp0*wp0*Cx[0] + wp1*wp1*Cx[3] + wp2*wp2*Cx[5]
             + 2.f*(wp0*wp1*Cx[1] + wp0*wp2*Cx[2] + wp1*wp2*Cx[4]);
  float sP = pg[o] * rsqrtf(varp + EPSF);
  float W0=sZ*wz0, W1=sZ*wz1, W2=sZ*wz2;
  float W4=sP*wp0, W5=sP*wp1, W6=sP*wp2;
  Wf[0*OO+o]=W0; Wf[1*OO+o]=W1; Wf[2*OO+o]=W2; Wf[3*OO+o]=0.f;
  Wf[4*OO+o]=W4; Wf[5*OO+o]=W5; Wf[6*OO+o]=W6; Wf[7*OO+o]=0.f;
  bias[o] = bb[o] - (W0*my[0]+W1*my[1]+W2*my[2])
          + pbeta[o] - (W4*mx[0]+W5*mx[1]+W6*mx[2]);
}

// ---- k8: fused epilogue GEMM via V_WMMA_F32_16X16X4_F32 --------------------
// D(16 o-chans x 16 positions) = A(W^T tile, 16x4) x B(features, 4x16), K=8 in
// two chained WMMAs. Output stored in (N,O,T,V) flat order (raw-reshape match);
// lanes 0-15 of each D VGPR are 16 consecutive positions -> coalesced stores.
__global__ void __launch_bounds__(128) k8_gemm(const float* __restrict__ y,
                                               const float* __restrict__ xf,
                                               const float* __restrict__ Wf,
                                               float* __restrict__ out) {
  __shared__ float Wl[8*OO + OO];          // fused W (512) + bias (64), contiguous in ws
  for (int t = threadIdx.x; t < 8*OO + OO; t += 128) Wl[t] = Wf[t];
  __syncthreads();
  int wave = threadIdx.x >> 5;
  int lane = threadIdx.x & 31;
  int lm   = lane & 15;
  bool lo16 = lane < 16;
  int pbase = (blockIdx.x * 4 + wave) * 16;      // 16 positions per wave
  int p = pbase + lm;                             // TV % 16 == 0: no n-crossing
  int n = p / TV, q = p % TV;
  int fb = n * (3*TV) + q;
  float y0 = y[fb], y1 = y[fb+TV], y2 = y[fb+2*TV];
  float x0 = xf[fb], x1 = xf[fb+TV], x2 = xf[fb+2*TV];
  // B fragment (4x16): VGPR0 = {K0 | K2}, VGPR1 = {K1 | K3} across lane halves
  v2f b_lo, b_hi;
  b_lo.x = lo16 ? y0 : y2;  b_lo.y = lo16 ? y1 : 0.f;
  b_hi.x = lo16 ? x0 : x2;  b_hi.y = lo16 ? x1 : 0.f;
  int k0  = lo16 ? 0 : 2;
  int hi8 = lo16 ? 0 : 8;
  int outbase = n * (OO*TV) + q;
  for (int ot = 0; ot < 4; ++ot) {               // 4 tiles of 16 output channels
    int oa = ot*16 + lm;
    // A fragment (16x4): lanes 0-15 hold K={0,1}, lanes 16-31 hold K={2,3}
    v2f a_lo, a_hi;
    a_lo.x = Wl[(k0    )*OO + oa];  a_lo.y = Wl[(k0 + 1)*OO + oa];
    a_hi.x = Wl[(k0 + 4)*OO + oa];  a_hi.y = Wl[(k0 + 5)*OO + oa];
    v8f acc = {};
    acc = __builtin_amdgcn_wmma_f32_16x16x4_f32(false, a_lo, false, b_lo,
                                                (short)0, acc, false, false);
    acc = __builtin_amdgcn_wmma_f32_16x16x4_f32(false, a_hi, false, b_hi,
                                                (short)0, acc, false, false);
    #pragma unroll
    for (int i = 0; i < 8; ++i) {                // D VGPR i -> rows i / i+8
      int o = ot*16 + i + hi8;
      float v = acc[i] + Wl[8*OO + o];
      out[outbase + o*TV] = fmaxf(v, 0.f);
    }
  }
}

extern "C" void kernel_launch(void* const* d_in, const int* in_sizes, int n_in,
                              void* d_out, int out_size, void* d_ws, size_t ws_size,
                              hipStream_t stream) {
  (void)in_sizes; (void)n_in; (void)out_size; (void)ws_size;
  const float* x     = (const float*)d_in[0];
  const float* w1    = (const float*)d_in[1];
  const float* b1    = (const float*)d_in[2];
  const float* w2    = (const float*)d_in[3];
  const float* b2    = (const float*)d_in[4];
  const float* cw    = (const float*)d_in[5];
  const float* cb    = (const float*)d_in[6];
  const float* cg    = (const float*)d_in[7];
  const float* cbeta = (const float*)d_in[8];
  const float* PA    = (const float*)d_in[9];
  const float* alpha = (const float*)d_in[10];
  const float* c2w   = (const float*)d_in[11];
  const float* c2b   = (const float*)d_in[12];
  const float* bng   = (const float*)d_in[13];
  const float* bnb   = (const float*)d_in[14];
  const float* pw    = (const float*)d_in[15];
  const float* pb    = (const float*)d_in[16];
  const float* pg    = (const float*)d_in[17];
  const float* pbeta = (const float*)d_in[18];

  float* ws    = (float*)d_ws;
  float* out   = (float*)d_out;
  float* a1    = ws + WS_A1;
  float* a2    = ws + WS_A2;
  float* y     = ws + WS_Y;
  float* AM    = ws + WS_AM;
  float* stats = ws + WS_ST;
  float* Wf    = ws + WS_W;
  float* bias  = ws + WS_BI;

  hipMemsetAsync(stats, 0, 64*sizeof(float), stream);
  k1_proj   <<<(NTV+255)/256, 256, 0, stream>>>(x, w1, b1, w2, b2, cw, cb, a1, a2);
  k2_sstats <<<NT/256,        256, 0, stream>>>(a1, a2, cb, stats);
  k3_final  <<<1,               1, 0, stream>>>(stats);
  k4_adj    <<<VVD*VVD,        64, 0, stream>>>(a1, a2, cb, cg, cbeta, stats, PA, alpha, AM);
  k5_y      <<<2048,          256, 0, stream>>>(x, AM, y);
  k6_moments<<<1024,          256, 0, stream>>>(y, x, stats);
  k7_fuse   <<<1,              64, 0, stream>>>(stats, c2w, c2b, bng, bnb, pw, pb, pg, pbeta, Wf, bias);
  k8_gemm   <<<NTV/64,        128, 0, stream>>>(y, x, Wf, out);
}